// Jacobi_55645596286989
// MI455X (gfx1250) — compile-verified
//
#include <hip/hip_runtime.h>

// Damped Jacobi sweep: out = x + (w/Ac) * (f - conv3x3_per_batch(x, kA))
// B=64, H=W=512, fp32. HBM-bound (~1.7 FLOP/byte) -> optimize data movement:
//  * async global->LDS DMA (CDNA5 ASYNCcnt path) to stage x tiles with halo
//  * full-width tiles: horizontal halo == zero image border (no extra traffic)
//  * non-temporal f loads / out stores so x stays resident in the 192MB L2
//  * uniform per-batch taps via scalar loads; v_fmac_f32 body

#define IMG_H 512
#define IMG_W 512
#define BATCH 64
#define TILE_H 8
#define LROWS (TILE_H + 2)          // tile rows incl. top/bottom halo
#define LDS_W 520                   // 4 pad | 512 | 4 pad  (keeps 16B alignment)
#define NTHREADS 256

typedef float v4f __attribute__((ext_vector_type(4)));
typedef int   v4i __attribute__((ext_vector_type(4)));

typedef __attribute__((address_space(1))) v4i* g1_v4i;  // global src (non-const!)
typedef __attribute__((address_space(3))) v4i* l3_v4i;  // LDS dst

__device__ __forceinline__ void async_copy_b128(const float* gp, float* lp) {
#if __has_builtin(__builtin_amdgcn_global_load_async_to_lds_b128)
  __builtin_amdgcn_global_load_async_to_lds_b128((g1_v4i)(float*)gp, (l3_v4i)lp, 0, 0);
#else
  // async encoding: VDST = LDS byte-address VGPR, VADDR = 64-bit global addr
  asm volatile("global_load_async_to_lds_b128 %0, %1, off"
               :: "v"((unsigned)(unsigned long long)lp),
                  "v"((unsigned long long)gp)
               : "memory");
#endif
}

__device__ __forceinline__ void wait_async_zero() {
#if __has_builtin(__builtin_amdgcn_s_wait_asynccnt)
  __builtin_amdgcn_s_wait_asynccnt(0);
#else
  asm volatile("s_wait_asynccnt 0" ::: "memory");
#endif
}

__global__ __launch_bounds__(NTHREADS) void
jacobi_stencil_kernel(const float* __restrict__ x,
                      const float* __restrict__ f,
                      const float* __restrict__ kA,
                      float* __restrict__ out)
{
  __shared__ float tile[LROWS * LDS_W];   // 20800 B

  const int tid = threadIdx.x;
  const int blk = blockIdx.x;
  const int b   = blk >> 6;               // 64 row-tiles per image
  const int ty0 = (blk & 63) * TILE_H;

  const float* xb = x   + (size_t)b * (IMG_H * IMG_W);
  const float* fb = f   + (size_t)b * (IMG_H * IMG_W);
  float*       ob = out + (size_t)b * (IMG_H * IMG_W);

  // 9 per-batch taps: uniform -> scalar loads
  const float* kb = kA + b * 9;
  const float k0 = kb[0], k1 = kb[1], k2 = kb[2];
  const float k3 = kb[3], k4 = kb[4], k5 = kb[5];
  const float k6 = kb[6], k7 = kb[7], k8 = kb[8];
  const float wsc = (2.0f / 3.0f) / k4;

  // zero column halo (image border): lds col 3 == x[-1], lds col 516 == x[512]
  if (tid < 2 * LROWS) {
    const int r = tid >> 1;
    const int c = (tid & 1) ? 3 : (4 + IMG_W);
    tile[r * LDS_W + c] = 0.0f;
  }

  // warm L2 for the f tile while the async DMA runs (one line per 128B)
  for (int t = tid; t < TILE_H * (IMG_W / 4); t += NTHREADS) {
    if ((t & 7) == 0) {
      const int r = t >> 7, q = t & 127;
      __builtin_prefetch(fb + (size_t)(ty0 + r) * IMG_W + q * 4, 0, 0);
    }
  }

  // stage x rows [ty0-1, ty0+TILE_H] into LDS: 16B async copies per lane
  for (int t = tid; t < LROWS * (IMG_W / 4); t += NTHREADS) {
    const int r  = t >> 7;                // IMG_W/4 == 128
    const int q  = t & 127;
    const int gy = ty0 - 1 + r;
    float* lp = &tile[r * LDS_W + 4 + q * 4];
    if (gy >= 0 && gy < IMG_H) {
      async_copy_b128(xb + (size_t)gy * IMG_W + q * 4, lp);
    } else {
      v4f z = {0.f, 0.f, 0.f, 0.f};       // zero-pad out-of-image halo row
      *(v4f*)lp = z;
    }
  }

  wait_async_zero();     // my DMA done before I signal
  __syncthreads();       // everyone's DMA + zero-fill visible

  // each thread produces float4 outputs: 3x6 LDS window -> 9-tap correlation
  for (int t = tid; t < TILE_H * (IMG_W / 4); t += NTHREADS) {
    const int r  = t >> 7;
    const int q  = t & 127;
    const int c0 = q * 4;
    const int gy = ty0 + r;

    const float* r0 = &tile[(r + 0) * LDS_W + 3 + c0];
    const float* r1 = &tile[(r + 1) * LDS_W + 3 + c0];
    const float* r2 = &tile[(r + 2) * LDS_W + 3 + c0];

    const float a0=r0[0], a1=r0[1], a2=r0[2], a3=r0[3], a4=r0[4], a5=r0[5];
    const float b0=r1[0], b1=r1[1], b2=r1[2], b3=r1[3], b4=r1[4], b5=r1[5];
    const float c0v=r2[0],c1=r2[1], c2=r2[2], c3=r2[3], c4=r2[4], c5=r2[5];

    const v4f fv =
        __builtin_nontemporal_load((const v4f*)(fb + (size_t)gy * IMG_W + c0));

    const float ax0 = k0*a0 + k1*a1 + k2*a2 + k3*b0 + k4*b1 + k5*b2 + k6*c0v + k7*c1 + k8*c2;
    const float ax1 = k0*a1 + k1*a2 + k2*a3 + k3*b1 + k4*b2 + k5*b3 + k6*c1  + k7*c2 + k8*c3;
    const float ax2 = k0*a2 + k1*a3 + k2*a4 + k3*b2 + k4*b3 + k5*b4 + k6*c2  + k7*c3 + k8*c4;
    const float ax3 = k0*a3 + k1*a4 + k2*a5 + k3*b3 + k4*b4 + k5*b5 + k6*c3  + k7*c4 + k8*c5;

    v4f o;
    o.x = b1 + wsc * (fv.x - ax0);
    o.y = b2 + wsc * (fv.y - ax1);
    o.z = b3 + wsc * (fv.z - ax2);
    o.w = b4 + wsc * (fv.w - ax3);

    __builtin_nontemporal_store(o, (v4f*)(ob + (size_t)gy * IMG_W + c0));
  }
}

extern "C" void kernel_launch(void* const* d_in, const int* in_sizes, int n_in,
                              void* d_out, int out_size, void* d_ws, size_t ws_size,
                              hipStream_t stream) {
  (void)in_sizes; (void)n_in; (void)out_size; (void)d_ws; (void)ws_size;
  const float* x  = (const float*)d_in[0];
  const float* f  = (const float*)d_in[1];
  const float* kA = (const float*)d_in[2];
  float* out = (float*)d_out;

  dim3 grid(BATCH * (IMG_H / TILE_H));   // 4096 blocks
  dim3 block(NTHREADS);                  // 8 waves of 32
  jacobi_stencil_kernel<<<grid, block, 0, stream>>>(x, f, kA, out);
}